// AttentionMapLayer_47339129536789
// MI455X (gfx1250) — compile-verified
//
#include <hip/hip_runtime.h>

typedef __attribute__((ext_vector_type(2))) float v2f;
typedef __attribute__((ext_vector_type(4))) float f4v;
typedef __attribute__((ext_vector_type(8))) float v8f;

#define HEIGHT   300
#define WIDTH    25
#define CH       16
#define HTILES   19            // ceil(300/16)
#define WAVES    8             // 256 threads / wave32
#define ROI_MASK 0xA5108u      // bits 3,8,12,14,17,19

// ---------------------------------------------------------------------------
// Kernel 1: per-batch-row inverse L2 norms (keras l2_normalize semantics:
// x * rsqrt(max(sum(x^2), 1e-12))). Tiny: ~1.3 MB read total.
// ---------------------------------------------------------------------------
__global__ __launch_bounds__(256) void norms_kernel(
    const float* __restrict__ s_o, const float* __restrict__ t_o,
    float* __restrict__ inv_s, float* __restrict__ inv_t, int B) {
  int b = blockIdx.x * blockDim.x + threadIdx.x;
  if (b >= B) return;
  float ss = 0.f;
  const float* sp = s_o + b * WIDTH;
  #pragma unroll
  for (int i = 0; i < WIDTH; ++i) { float v = sp[i]; ss += v * v; }
  inv_s[b] = rsqrtf(fmaxf(ss, 1e-12f));
  float st = 0.f;
  const float* tp = t_o + b * HEIGHT;
  for (int i = 0; i < HEIGHT; ++i) { float v = tp[i]; st += v * v; }
  inv_t[b] = rsqrtf(fmaxf(st, 1e-12f));
}

// ---------------------------------------------------------------------------
// Kernel 2: one wave per (b, 16-row h-tile).
// Stage 1: a-tile = t_hat (16x1) x s_hat (1x16) + roi via V_WMMA_F32_16X16X4_F32
//          (rank-1: only K=0 populated; C carries the roi tile). Two WMMAs
//          cover w in [0,32); results stored CELL-MAJOR (a[m*25+w]) in wave LDS
//          so the stream loop indexes it with immediate offsets.
// Stage 2: stream the wave's contiguous 25.6KB ipt region with NT b128
//          loads/stores, scaling by a[cell]. float4 index = lane + 32*i.
// ---------------------------------------------------------------------------
__global__ __launch_bounds__(256) void amap_kernel(
    const float* __restrict__ s_o, const float* __restrict__ t_o,
    const float* __restrict__ ipt,
    const float* __restrict__ inv_s, const float* __restrict__ inv_t,
    float* __restrict__ out, int ntiles) {
  __shared__ float lds_a[WAVES][16 * WIDTH];   // 1.6 KB per wave, cell-major

  const int wave = threadIdx.x >> 5;
  const int lane = threadIdx.x & 31;
  const int tile = blockIdx.x * WAVES + wave;
  if (tile >= ntiles) return;                 // wave-uniform

  const int b  = tile / HTILES;
  const int h0 = (tile % HTILES) * 16;

  const float ivt = inv_t[b];
  const float ivs = inv_s[b];

  const int  l16     = lane & 15;
  const bool lowhalf = lane < 16;

  // A-matrix (16x4 f32): lanes 0-15 hold {K=0,K=1}, lanes 16-31 hold {K=2,K=3}.
  // Rank-1 product => only K=0 nonzero.
  float tval = 0.f;
  if (lowhalf) {
    int hh = h0 + l16;
    if (hh < HEIGHT) tval = t_o[b * HEIGHT + hh] * ivt;
  }
  v2f A; A.x = tval; A.y = 0.f;

  float* my = lds_a[wave];
  const int mbase = lowhalf ? 0 : 8;          // C/D layout: lanes>=16 carry M+8

  #pragma unroll
  for (int w0 = 0; w0 < 32; w0 += 16) {
    const int w = w0 + l16;                   // N index this lane carries
    float sval = 0.f;
    if (lowhalf && w < WIDTH) sval = s_o[b * WIDTH + w] * ivs;
    v2f Bv; Bv.x = sval; Bv.y = 0.f;

    // C = roi tile: C[m,n] = roi(w0+n), constant down columns.
    float rv = ((ROI_MASK >> w) & 1u) ? 1.0f : 0.0f;
    v8f C = {rv, rv, rv, rv, rv, rv, rv, rv};

    // D = A x B + C  ==  t_hat x s_hat^T + roi   (exactly the reference's `a`)
    v8f D = __builtin_amdgcn_wmma_f32_16x16x4_f32(
        /*neg_a=*/false, A, /*neg_b=*/false, Bv,
        /*c_mod=*/(short)0, C, /*reuse_a=*/false, /*reuse_b=*/false);

    if (w < WIDTH) {                          // mask w>=25 (cell-major layout)
      #pragma unroll
      for (int r = 0; r < 8; ++r)
        my[(mbase + r) * WIDTH + w] = D[r];   // row M=r(+8), col N
    }
  }
  __builtin_amdgcn_wave_barrier();   // wave-private LDS; per-wave DS is in-order

  // ---- Stage 2: stream 16 x 25 x 16 floats (contiguous, h-major) ----------
  // lane = slot*4 + q  =>  float4 index within tile = lane + 32*i;
  // cell index for this lane = slot + 8*i  =>  a addr = slot*4 + 32*i bytes.
  const int slot = lane >> 2;

  const size_t base = ((size_t)b * HEIGHT + h0) * (WIDTH * CH);
  const f4v* __restrict__ ip = (const f4v*)(ipt + base);
  f4v* __restrict__       op = (f4v*)(out + base);
  const float* __restrict__ ap = my + slot;

  if (h0 + 16 <= HEIGHT) {
    // Fast path (18 of 19 tiles): no guards, immediate offsets, batched loads.
    #pragma unroll 10
    for (int i = 0; i < (16 * WIDTH * CH) / (4 * 32); ++i) {   // 50 iters
      float a = ap[8 * i];
      f4v v = __builtin_nontemporal_load(ip + lane + 32 * i);
      __builtin_nontemporal_store(v * a, op + lane + 32 * i);
    }
  } else {
    // Tail tile (h0 = 288): cells valid while cell < (HEIGHT-h0)*WIDTH.
    const int limit = (HEIGHT - h0) * WIDTH;
    for (int i = 0; i < (16 * WIDTH * CH) / (4 * 32); ++i) {
      int cell = slot + 8 * i;
      if (cell >= limit) break;               // monotone -> clean exit
      float a = my[cell];
      f4v v = __builtin_nontemporal_load(ip + lane + 32 * i);
      __builtin_nontemporal_store(v * a, op + lane + 32 * i);
    }
  }
}

// ---------------------------------------------------------------------------
extern "C" void kernel_launch(void* const* d_in, const int* in_sizes, int n_in,
                              void* d_out, int out_size, void* d_ws, size_t ws_size,
                              hipStream_t stream) {
  const float* s_o = (const float*)d_in[0];   // [B, 25]
  const float* t_o = (const float*)d_in[1];   // [B, 300]
  const float* ipt = (const float*)d_in[2];   // [B, 300, 25, 16]
  float*       out = (float*)d_out;           // [B, 300, 25, 16]

  const int B = in_sizes[0] / WIDTH;

  float* inv_s = (float*)d_ws;                // B floats
  float* inv_t = inv_s + B;                   // B floats  (8 KB total @ B=1024)

  norms_kernel<<<(B + 255) / 256, 256, 0, stream>>>(s_o, t_o, inv_s, inv_t, B);

  const int ntiles  = B * HTILES;             // 19456 waves @ B=1024
  const int nblocks = (ntiles + WAVES - 1) / WAVES;
  amap_kernel<<<nblocks, 256, 0, stream>>>(s_o, t_o, ipt, inv_s, inv_t, out, ntiles);
}